// BCNLayer_16638703304941
// MI455X (gfx1250) — compile-verified
//
#include <hip/hip_runtime.h>

#ifndef __has_builtin
#define __has_builtin(x) 0
#endif

#if __has_builtin(__builtin_amdgcn_global_load_async_to_lds_b128)
#define HAVE_ASYNC_LDS 1
#else
#define HAVE_ASYNC_LDS 0
#endif

#if HAVE_ASYNC_LDS
#if __has_builtin(__builtin_amdgcn_s_wait_asynccnt)
#define WAIT_ASYNC() __builtin_amdgcn_s_wait_asynccnt(0)
#else
#define WAIT_ASYNC() asm volatile("s_wait_asynccnt 0" ::: "memory")
#endif
#else
#define WAIT_ASYNC() ((void)0)
#endif

namespace {

constexpr int LH = 128;           // lattice height
constexpr int LW = 128;           // lattice width
constexpr int NHW = LH * LW;      // 16384
constexpr int NB = 1024;          // batch
constexpr int JT = 32;            // output cols per workgroup (one per lane)
constexpr int IC = 16;            // output rows per workgroup chunk
constexpr int BT = 64;            // batch per workgroup
constexpr int VB = 8;             // batch per thread
constexpr int SRCC = JT + 6;      // 38 staged source columns (halo +-3)
constexpr int XPITCH = 68;        // padded batch pitch in floats: 16B-aligned, 4-bank lane stride

typedef int int4v __attribute__((ext_vector_type(4)));
typedef float v4f __attribute__((ext_vector_type(4)));
typedef float v2f __attribute__((ext_vector_type(2)));
typedef __attribute__((address_space(1))) int4v gint4_t;
typedef __attribute__((address_space(3))) int4v lint4_t;

__device__ __forceinline__ float sigf(float v) {
  return 1.0f / (1.0f + __expf(-v));
}

// Stage one source row (38 cols x 64 batch floats) into LDS, batch-coalesced.
// Out-of-lattice columns are zero-filled so compute needs no masking.
__device__ __forceinline__ void stage_row(float* __restrict__ dst,
                                          const float* __restrict__ X,
                                          int row, int j0, int b0, int flat) {
  for (int c = flat; c < SRCC * 16; c += 256) {   // 608 chunks of 16B
    const int cc = c >> 4;          // staged column index 0..37
    const int bq = (c & 15) << 2;   // batch offset within tile (floats)
    const int srcj = j0 - 3 + cc;
    float* ld = dst + cc * XPITCH + bq;           // (cc*68+bq)*4 is 16B aligned
    if (srcj >= 0 && srcj < LW) {
      const float* g = X + (size_t)(row * LW + srcj) * NB + b0 + bq;
#if HAVE_ASYNC_LDS
      __builtin_amdgcn_global_load_async_to_lds_b128((gint4_t*)g, (lint4_t*)ld, 0, 0);
#else
      *(v4f*)ld = *(const v4f*)g;
#endif
    } else {
      *(v4f*)ld = (v4f)0.f;
    }
  }
}

__global__ __launch_bounds__(256)
void bcn_stencil_kernel(const float* __restrict__ X,
                        const float* __restrict__ Wg,
                        float* __restrict__ Y) {
  __shared__ float xstage[2][SRCC * XPITCH];  // double-buffered x row tiles
  __shared__ float xfer[2][JT * XPITCH];      // output transpose buffers

  const int lane = threadIdx.x;               // output column within tile
  const int ty = threadIdx.y;                 // batch chunk (VB each)
  const int flat = ty * 32 + lane;

  const int bid = blockIdx.x;
  const int j0 = (bid & 3) * JT;              // 4 column tiles
  const int i0 = ((bid >> 2) & 7) * IC;       // 8 row chunks
  const int b0 = (bid >> 5) * BT;             // 16 batch tiles

  const int ocol = j0 + lane;

  // Rolling accumulator window: acc[k][h] accumulates output row (r + k - 3),
  // batch pair h. v2f accumulators target v_pk_fma_f32 (2 FMA/lane/instr).
  v2f acc[7][4];
#pragma unroll
  for (int k = 0; k < 7; ++k)
#pragma unroll
    for (int h = 0; h < 4; ++h) acc[k][h] = (v2f)0.f;

  const int rstart = i0 - 3;
  const int rlast = i0 + IC + 2;              // last source row that contributes

  if (rstart >= 0) stage_row(xstage[0], X, rstart, j0, b0, flat);
  WAIT_ASYNC();
  __syncthreads();

  int pb = 0;   // x buffer parity
  int q = 0;    // xfer buffer parity
  for (int r = rstart; r <= rlast + 1; ++r) {
    // (1) async-stage next source row while computing this one
    const int rn = r + 1;
    if (rn >= 0 && rn < LH && rn <= rlast)
      stage_row(xstage[pb ^ 1], X, rn, j0, b0, flat);

    // (2) coalesced global store of output row finalized last iteration
    const int rs = r - 4;
    if (rs >= i0) {
      const float* sb = xfer[q ^ 1];
      const int scol = flat >> 3;             // 0..31
      const int brel = (flat & 7) * 8;        // 0..56
      const v4f v0 = *(const v4f*)&sb[scol * XPITCH + brel];
      const v4f v1 = *(const v4f*)&sb[scol * XPITCH + brel + 4];
      float* g = Y + (size_t)(rs * LW + j0 + scol) * NB + b0 + brel;
      *(v4f*)g = v0;
      *(v4f*)(g + 4) = v1;
    }

    // (3) accumulate all 49 taps sourced from row r
    if (r >= 0 && r < LH && r <= rlast) {
      const float* xb = xstage[pb];
      v2f xv[7][4];
#pragma unroll
      for (int d = 0; d < 7; ++d) {           // d = dx+3; src col = ocol+3-d
        const int idx = (lane + 6 - d) * XPITCH + ty * VB;
        const v4f ta = *(const v4f*)&xb[idx];
        const v4f tb = *(const v4f*)&xb[idx + 4];
        xv[d][0] = __builtin_shufflevector(ta, ta, 0, 1);
        xv[d][1] = __builtin_shufflevector(ta, ta, 2, 3);
        xv[d][2] = __builtin_shufflevector(tb, tb, 0, 1);
        xv[d][3] = __builtin_shufflevector(tb, tb, 2, 3);
      }
      int wq[7];
#pragma unroll
      for (int d = 0; d < 7; ++d) {
        int cs = ocol + 3 - d;
        cs = cs < 0 ? 0 : (cs > LW - 1 ? LW - 1 : cs);  // clamped addr; x is 0 there
        wq[d] = r * LW + cs;
      }
#pragma unroll
      for (int dy = 0; dy < 7; ++dy) {        // output row = r + dy - 3
        const int orow = r + dy - 3;
        if (orow < i0 || orow >= i0 + IC) continue;     // wave-uniform branch
#pragma unroll
        for (int d = 0; d < 7; ++d) {
          const float wv = Wg[(size_t)(dy * 7 + d) * NHW + wq[d]];
          const v2f w2 = (v2f)wv;             // splat
#pragma unroll
          for (int h = 0; h < 4; ++h)
            acc[dy][h] = __builtin_elementwise_fma(xv[d][h], w2, acc[dy][h]);
        }
      }
    }

    // (4) output row r-3 is complete: sigmoid and park in transpose buffer
    const int fi = r - 3;
    if (fi >= i0 && fi < i0 + IC) {
      float* db = &xfer[q][lane * XPITCH + ty * VB];
      db[0] = sigf(acc[0][0].x);
      db[1] = sigf(acc[0][0].y);
      db[2] = sigf(acc[0][1].x);
      db[3] = sigf(acc[0][1].y);
      db[4] = sigf(acc[0][2].x);
      db[5] = sigf(acc[0][2].y);
      db[6] = sigf(acc[0][3].x);
      db[7] = sigf(acc[0][3].y);
    }

    // (5) slide the accumulator window
#pragma unroll
    for (int k = 0; k < 6; ++k)
#pragma unroll
      for (int h = 0; h < 4; ++h) acc[k][h] = acc[k + 1][h];
#pragma unroll
    for (int h = 0; h < 4; ++h) acc[6][h] = (v2f)0.f;

    WAIT_ASYNC();
    __syncthreads();
    pb ^= 1;
    q ^= 1;
  }
}

} // namespace

extern "C" void kernel_launch(void* const* d_in, const int* in_sizes, int n_in,
                              void* d_out, int out_size, void* d_ws, size_t ws_size,
                              hipStream_t stream) {
  (void)in_sizes; (void)n_in; (void)d_ws; (void)ws_size; (void)out_size;
  const float* X = (const float*)d_in[0];    // (16384, 1024) f32
  const float* Wg = (const float*)d_in[1];   // (7, 7, 16384) f32
  float* Y = (float*)d_out;                  // (16384, 1024) f32

  dim3 block(32, 8);
  dim3 grid(4 * 8 * 16);                     // j-tiles * i-chunks * b-tiles = 512
  bcn_stencil_kernel<<<grid, block, 0, stream>>>(X, Wg, Y);
}